// Memory_cell2_6957847019563
// MI455X (gfx1250) — compile-verified
//
#include <hip/hip_runtime.h>

typedef __attribute__((ext_vector_type(2))) float v2f;
typedef __attribute__((ext_vector_type(8))) float v8f;

#define N_ROWS 8192
#define R_DIM  512
#define K_DIM  20000
#define ROWS   32          // h rows per workgroup
#define KC     128         // K columns per chunk (16 per wave, 8 waves)
#define SHS    516         // LDS stride for h tile (pad: bank-conflict free, 16B aligned)
#define ATS    132         // LDS stride for att tile

__global__ __launch_bounds__(256, 1)
void memcell_fused_softmax_wmma(const float* __restrict__ h,
                                const float* __restrict__ M,
                                float* __restrict__ out)
{
    __shared__ float sh [ROWS * SHS];   // h tile (32 x 512, padded)
    __shared__ float att[ROWS * ATS];   // logits / probs tile (32 x 128, padded)
    __shared__ float pmax[ROWS * 8];
    __shared__ float psum[ROWS * 8];
    __shared__ float mrow[ROWS];
    __shared__ float lrow[ROWS];
    __shared__ float cfac[ROWS];

    const int tid  = threadIdx.x;
    const int wave = tid >> 5;
    const int lane = tid & 31;
    const int lo   = lane & 15;
    const int hi   = lane >> 4;
    const int r0   = blockIdx.x * ROWS;

    // ---- preload h tile into LDS (float2 granularity) ----
    for (int i = tid; i < ROWS * (R_DIM / 2); i += 256) {
        int row = i >> 8;          // 256 float2 per row
        int c2  = (i & 255) << 1;
        v2f v = *(const v2f*)(h + (size_t)(r0 + row) * R_DIM + c2);
        *(v2f*)&sh[row * SHS + c2] = v;
    }
    if (tid < ROWS) { mrow[tid] = -3.0e38f; lrow[tid] = 0.0f; }
    __syncthreads();

    // out accumulators: this wave owns columns [wave*64, wave*64+64) for both 16-row stripes
    v8f acc[2][4];
#pragma unroll
    for (int s = 0; s < 2; s++)
#pragma unroll
        for (int ct = 0; ct < 4; ct++) acc[s][ct] = 0.0f;

    const int srow = tid & 31;   // stats-pass row
    const int sseg = tid >> 5;   // stats-pass 16-col segment

    for (int kc = 0; kc < K_DIM; kc += KC) {
        // ================= phase 1: logits S = h_tile @ M_chunk^T =================
        const int kcol = kc + (wave << 4) + lo;                   // this lane's K column (B matrix)
        const int kcl  = (kcol < K_DIM) ? kcol : (K_DIM - 1);     // clamp OOB rows of M
        const float* bp  = M + (size_t)kcl * R_DIM + (hi << 1);
        const float* a0p = &sh[lo * SHS + (hi << 1)];
        const float* a1p = &sh[(lo + 16) * SHS + (hi << 1)];
        v8f s0 = 0.0f, s1 = 0.0f;
#pragma unroll 4
        for (int rk = 0; rk < R_DIM; rk += 4) {
            v2f a0 = *(const v2f*)(a0p + rk);
            v2f a1 = *(const v2f*)(a1p + rk);
            v2f b  = *(const v2f*)(bp + rk);
            s0 = __builtin_amdgcn_wmma_f32_16x16x4_f32(false, a0, false, b, (short)0, s0, false, false);
            s1 = __builtin_amdgcn_wmma_f32_16x16x4_f32(false, a1, false, b, (short)0, s1, false, false);
        }
        if (kcol >= K_DIM) {   // padded K columns -> -inf logits -> zero softmax weight
#pragma unroll
            for (int i = 0; i < 8; i++) { s0[i] = -3.0e38f; s1[i] = -3.0e38f; }
        }
        {
            const int col = (wave << 4) + lo;
#pragma unroll
            for (int i = 0; i < 8; i++) {
                att[(i + (hi << 3)) * ATS + col]        = s0[i];
                att[(16 + i + (hi << 3)) * ATS + col]   = s1[i];
            }
        }
        __syncthreads();

        // ================= online softmax update =================
        {   // per-(row,segment) max
            const float* ar = &att[srow * ATS + (sseg << 4)];
            float mx = ar[0];
#pragma unroll
            for (int j = 1; j < 16; j++) mx = fmaxf(mx, ar[j]);
            pmax[(srow << 3) + sseg] = mx;
        }
        __syncthreads();
        if (tid < ROWS) {
            float mo = mrow[tid];
            float cm = pmax[tid << 3];
#pragma unroll
            for (int s = 1; s < 8; s++) cm = fmaxf(cm, pmax[(tid << 3) + s]);
            float mn = fmaxf(mo, cm);
            mrow[tid] = mn;
            cfac[tid] = __expf(mo - mn);
        }
        __syncthreads();
        {   // exponentiate in place, partial row sums
            const float mn = mrow[srow];
            float* aw = &att[srow * ATS + (sseg << 4)];
            float ss = 0.0f;
#pragma unroll
            for (int j = 0; j < 16; j++) { float p = __expf(aw[j] - mn); aw[j] = p; ss += p; }
            psum[(srow << 3) + sseg] = ss;
        }
        __syncthreads();
        if (tid < ROWS) {
            float t = 0.0f;
#pragma unroll
            for (int s = 0; s < 8; s++) t += psum[(tid << 3) + s];
            lrow[tid] = lrow[tid] * cfac[tid] + t;
        }

        // ================= phase 2: acc = acc*c + P @ M_chunk =================
#pragma unroll
        for (int s = 0; s < 2; s++) {
#pragma unroll
            for (int i = 0; i < 8; i++) {
                float f = cfac[(s << 4) + i + (hi << 3)];
#pragma unroll
                for (int ct = 0; ct < 4; ct++) acc[s][ct][i] *= f;
            }
        }
#pragma unroll
        for (int ct = 0; ct < 4; ct++) {
            const int ccol = (wave << 6) + (ct << 4) + lo;        // this lane's output column
#pragma unroll 4
            for (int kk = 0; kk < KC; kk += 4) {
                int kr  = kc + kk + (hi << 1);
                int kr0 = (kr     < K_DIM) ? kr     : (K_DIM - 1);
                int kr1 = (kr + 1 < K_DIM) ? kr + 1 : (K_DIM - 1);
                v2f b;
                b.x = M[(size_t)kr0 * R_DIM + ccol];
                b.y = M[(size_t)kr1 * R_DIM + ccol];
                v2f a0 = *(const v2f*)&att[lo * ATS + kk + (hi << 1)];
                v2f a1 = *(const v2f*)&att[(lo + 16) * ATS + kk + (hi << 1)];
                acc[0][ct] = __builtin_amdgcn_wmma_f32_16x16x4_f32(false, a0, false, b, (short)0, acc[0][ct], false, false);
                acc[1][ct] = __builtin_amdgcn_wmma_f32_16x16x4_f32(false, a1, false, b, (short)0, acc[1][ct], false, false);
            }
        }
        __syncthreads();   // protect att/cfac for next chunk
    }

    // ================= finalize: out = acc / l =================
#pragma unroll
    for (int s = 0; s < 2; s++) {
#pragma unroll
        for (int i = 0; i < 8; i++) {
            const int row = (s << 4) + i + (hi << 3);
            const float inv = 1.0f / lrow[row];
            const size_t gb = (size_t)(r0 + row) * R_DIM + (wave << 6) + lo;
#pragma unroll
            for (int ct = 0; ct < 4; ct++) out[gb + (ct << 4)] = acc[s][ct][i] * inv;
        }
    }
}

extern "C" void kernel_launch(void* const* d_in, const int* in_sizes, int n_in,
                              void* d_out, int out_size, void* d_ws, size_t ws_size,
                              hipStream_t stream) {
    (void)in_sizes; (void)n_in; (void)d_ws; (void)ws_size; (void)out_size;
    const float* h = (const float*)d_in[0];
    const float* M = (const float*)d_in[1];
    float* out = (float*)d_out;
    dim3 grid(N_ROWS / ROWS);   // 256 workgroups
    dim3 block(256);            // 8 waves (wave32)
    memcell_fused_softmax_wmma<<<grid, block, 0, stream>>>(h, M, out);
}